// MonoRGCN_88304527606397
// MI455X (gfx1250) — compile-verified
//
#include <hip/hip_runtime.h>
#include <hip/hip_bf16.h>
#include <math.h>

// ---------------- problem constants (from reference) ----------------
#define NN   10000
#define EE   320000
#define FDIM 256          // F_IN == HID == 256
#define NREL 2
#define NBAS 10

// ---------------- WMMA vector types ----------------
typedef __attribute__((ext_vector_type(2))) float v2f;
typedef __attribute__((ext_vector_type(8))) float v8f;

// =====================================================================
// zero fill
// =====================================================================
__global__ void rgcn_zero_f(float* __restrict__ p, long n) {
    long i = (long)blockIdx.x * blockDim.x + threadIdx.x;
    if (i < n) p[i] = 0.0f;
}

// =====================================================================
// per-(relation,dst) edge counts (graph structure is layer-invariant)
// =====================================================================
__global__ void rgcn_count(const int* __restrict__ ei, const int* __restrict__ et,
                           float* __restrict__ cnt) {
    int e = blockIdx.x * blockDim.x + threadIdx.x;
    if (e >= EE) return;
    int d = ei[EE + e];
    int r = et[e];
    atomicAdd(&cnt[(size_t)r * NN + d], 1.0f);
}

__global__ void rgcn_inv(const float* __restrict__ cnt, float* __restrict__ inv) {
    int i = blockIdx.x * blockDim.x + threadIdx.x;
    if (i < NREL * NN) inv[i] = 1.0f / fmaxf(cnt[i], 1.0f);
}

// =====================================================================
// edge scatter: h[r][dst][:] += x[src][:]   (one wave32 per edge)
// =====================================================================
__global__ __launch_bounds__(256) void rgcn_aggregate(
    const float* __restrict__ x, const int* __restrict__ ei,
    const int* __restrict__ et, float* __restrict__ h) {
    int e    = (int)((blockIdx.x * 256u + threadIdx.x) >> 5);
    int lane = threadIdx.x & 31;
    if (e >= EE) return;
    int s = ei[e];
    int d = ei[EE + e];
    int r = et[e];
    const float* xs = x + (size_t)s * FDIM;
    float*       hd = h + ((size_t)r * NN + d) * FDIM;
    __builtin_prefetch(xs, 0, 1);            // -> global_prefetch_b8
    #pragma unroll
    for (int c = 0; c < FDIM; c += 32)
        atomicAdd(&hd[c + lane], xs[c + lane]);   // global_atomic_add_f32 (L2 resident)
}

// =====================================================================
// build Wcat in k-PAIR-INTERLEAVED layout:
//   Wc[(k/2) * (2*dpad) + 2*o + (k&1)]   for k in [0,768)
//   rows:  [root ; comp[0]@basis ; comp[1]@basis], zero-padded cols
// =====================================================================
__global__ void rgcn_build_w(const float* __restrict__ basis, const float* __restrict__ comp,
                             const float* __restrict__ root, float* __restrict__ Wc,
                             int dout, int dpad) {
    long idx = (long)blockIdx.x * blockDim.x + threadIdx.x;
    long total = (long)3 * FDIM * dpad;
    if (idx >= total) return;
    int k = (int)(idx / dpad);
    int o = (int)(idx % dpad);
    float val = 0.0f;
    if (o < dout) {
        if (k < FDIM) {
            val = root[(size_t)k * dout + o];
        } else {
            int r = (k - FDIM) >> 8;            // relation 0/1
            int i = (k - FDIM) & 255;           // input feature
            const float* cp = comp + r * NBAS;
            float s = 0.0f;
            #pragma unroll
            for (int b = 0; b < NBAS; ++b)
                s += cp[b] * basis[((size_t)b * FDIM + i) * dout + o];
            val = s;
        }
    }
    Wc[(size_t)(k >> 1) * (2 * dpad) + 2 * o + (k & 1)] = val;
}

// =====================================================================
// fused layer GEMM:  out = [X | h0/c0 | h1/c1] @ Wcat + bias  (+ReLU)
//   * block tile 256x64, K pipelined in 32-wide LDS chunks
//   * per-wave 32x64 tile: 2 A-frags reuse 4 B-frags -> 8 wmma / K-step
//   * B pair-interleaved in LDS: every fragment = one aligned ds_load_b64
//   * 1/cnt row scaling fused into A staging
//   * fp32 WMMA 16x16x4 keeps exact fp32 semantics
// =====================================================================
#define TM  256
#define TN  64
#define KC  32
#define KT  (3 * FDIM)    // 768
#define LDA 34            // even -> b64-aligned A frags; 34*r mod 64 distinct for r<16
#define SBP 160           // pair-row stride (words); 160 mod 64 == 32 -> halves on disjoint banks

__global__ __launch_bounds__(256) void rgcn_gemm_wmma(
    const float* __restrict__ X,     // [M x 256]
    const float* __restrict__ H,     // [2 x NN x 256]
    const float* __restrict__ inv,   // [2 x NN]
    const float* __restrict__ Wc,    // [384 x 2*dpad] pair-interleaved
    const float* __restrict__ bias,  // [dout]
    float* __restrict__ out,         // [M x dout]
    int M, int dout, int dpad, int relu)
{
    __shared__ float sA[TM * LDA];        // 256 x 32 (+pad)           = 34816 B
    __shared__ float sB[(KC / 2) * SBP];  // 16 pair-rows x 128 (+pad) = 10240 B

    const int tid  = threadIdx.x;
    const int lane = tid & 31;
    const int wave = tid >> 5;                // 0..7
    const int m0   = blockIdx.x * TM;
    const int n0   = blockIdx.y * TN;
    const int wrow = wave * 32;               // wave's 32-row strip

    v8f acc[8] = {{}, {}, {}, {}, {}, {}, {}, {}};   // [rowgrp 0/1][colgrp 0..3]

    float4 ra[8];   // staged A chunk: 256 rows x 32 floats / 256 thr = 8 f4
    float4 rb[2];   // staged B chunk: 16 pair-rows x 128 words / 256 thr = 2 f4

    auto fetch = [&](int kc) {                // global -> regs (row-scaled)
        const int s  = kc >> 8;               // 0: X, 1: h0, 2: h1
        const int ko = kc & 255;
        const float* Abase = (s == 0) ? X : (H + (size_t)(s - 1) * NN * FDIM);
        const float* invr  = inv + (size_t)(s - 1) * NN;
        #pragma unroll
        for (int i = 0; i < 8; ++i) {
            int idx = tid + i * 256;          // 0..2047
            int r   = idx >> 3;
            int c4  = (idx & 7) << 2;
            int gr  = m0 + r;
            float4 v = make_float4(0.f, 0.f, 0.f, 0.f);
            if (gr < M) {
                v = *(const float4*)(Abase + (size_t)gr * FDIM + ko + c4);
                if (s) {                      // fused mean-normalization
                    float sc = invr[gr];
                    v.x *= sc; v.y *= sc; v.z *= sc; v.w *= sc;
                }
            }
            ra[i] = v;
        }
        const int kcp = kc >> 1;              // pair-row base in Wcat
        #pragma unroll
        for (int i = 0; i < 2; ++i) {
            int idx = tid + i * 256;          // 0..511
            int p   = idx >> 5;               // pair-row 0..15
            int c4  = (idx & 31) << 2;        // word in 128-word row
            rb[i] = *(const float4*)(Wc + (size_t)(kcp + p) * (2 * dpad) + n0 * 2 + c4);
        }
    };
    auto commit = [&]() {                     // regs -> LDS
        #pragma unroll
        for (int i = 0; i < 8; ++i) {
            int idx = tid + i * 256;
            float* d = &sA[(idx >> 3) * LDA + ((idx & 7) << 2)];
            d[0] = ra[i].x; d[1] = ra[i].y; d[2] = ra[i].z; d[3] = ra[i].w;
        }
        #pragma unroll
        for (int i = 0; i < 2; ++i) {
            int idx = tid + i * 256;
            float* d = &sB[(idx >> 5) * SBP + ((idx & 31) << 2)];
            d[0] = rb[i].x; d[1] = rb[i].y; d[2] = rb[i].z; d[3] = rb[i].w;
        }
    };

    fetch(0);
    commit();
    __syncthreads();

    for (int kc = 0; kc < KT; kc += KC) {
        if (kc + KC < KT) fetch(kc + KC);     // overlap next chunk with compute

        const int half = (lane >> 4);         // 0 (lanes<16) / 1 (lanes>=16)
        const int rlo  = wrow + (lane & 15);
        const int rhi  = rlo + 16;
        const int col  = lane & 15;
        #pragma unroll
        for (int kk = 0; kk < KC; kk += 4) {
            int kA = kk + half * 2;           // A: k contiguous in row-major tile
            v2f a0 = *(const v2f*)(&sA[rlo * LDA + kA]);
            v2f a1 = *(const v2f*)(&sA[rhi * LDA + kA]);
            int pB = (kk >> 1) + half;        // B: pair-interleaved rows
            const float* bp = &sB[pB * SBP + 2 * col];
            v2f b0 = *(const v2f*)(bp);
            v2f b1 = *(const v2f*)(bp + 32);
            v2f b2 = *(const v2f*)(bp + 64);
            v2f b3 = *(const v2f*)(bp + 96);
            acc[0] = __builtin_amdgcn_wmma_f32_16x16x4_f32(false, a0, false, b0, (short)0, acc[0], false, false);
            acc[1] = __builtin_amdgcn_wmma_f32_16x16x4_f32(false, a0, false, b1, (short)0, acc[1], false, false);
            acc[2] = __builtin_amdgcn_wmma_f32_16x16x4_f32(false, a0, false, b2, (short)0, acc[2], false, false);
            acc[3] = __builtin_amdgcn_wmma_f32_16x16x4_f32(false, a0, false, b3, (short)0, acc[3], false, false);
            acc[4] = __builtin_amdgcn_wmma_f32_16x16x4_f32(false, a1, false, b0, (short)0, acc[4], false, false);
            acc[5] = __builtin_amdgcn_wmma_f32_16x16x4_f32(false, a1, false, b1, (short)0, acc[5], false, false);
            acc[6] = __builtin_amdgcn_wmma_f32_16x16x4_f32(false, a1, false, b2, (short)0, acc[6], false, false);
            acc[7] = __builtin_amdgcn_wmma_f32_16x16x4_f32(false, a1, false, b3, (short)0, acc[7], false, false);
        }

        __syncthreads();
        if (kc + KC < KT) {
            commit();
            __syncthreads();
        }
    }

    // ---- epilogue: bias + optional ReLU, ISA C/D layout ----
    const int colBase = n0 + (lane & 15);
    const int rowHalf = (lane >> 4) << 3;             // 0 or 8
    auto store_frag = [&](v8f av, int g, int c) {
        int col = colBase + c * 16;
        if (col >= dout) return;
        float bv = bias[col];
        #pragma unroll
        for (int v = 0; v < 8; ++v) {
            int row = m0 + wrow + g * 16 + rowHalf + v;
            if (row < M) {
                float val = av[v] + bv;
                if (relu) val = fmaxf(val, 0.0f);
                out[(size_t)row * dout + col] = val;
            }
        }
    };
    #pragma unroll
    for (int g = 0; g < 2; ++g)
        #pragma unroll
        for (int c = 0; c < 4; ++c)
            store_frag(acc[g * 4 + c], g, c);
}

// =====================================================================
// log_softmax over 40 classes (one wave32 per row)
// =====================================================================
__global__ __launch_bounds__(256) void rgcn_logsoftmax(const float* __restrict__ logits,
                                                       float* __restrict__ out) {
    int row  = blockIdx.x * 8 + (threadIdx.x >> 5);
    int lane = threadIdx.x & 31;
    if (row >= NN) return;
    const float* l = logits + (size_t)row * 40;
    float a = l[lane];                                  // cols 0..31
    float b = (lane < 8) ? l[32 + lane] : -3.4e38f;     // cols 32..39
    float m = fmaxf(a, b);
    #pragma unroll
    for (int o = 16; o > 0; o >>= 1) m = fmaxf(m, __shfl_xor(m, o, 32));
    float ea = expf(a - m);
    float eb = (lane < 8) ? expf(b - m) : 0.0f;
    float s  = ea + eb;
    #pragma unroll
    for (int o = 16; o > 0; o >>= 1) s += __shfl_xor(s, o, 32);
    float ls = logf(s);
    out[(size_t)row * 40 + lane] = a - m - ls;
    if (lane < 8) out[(size_t)row * 40 + 32 + lane] = b - m - ls;
}

// =====================================================================
// driver
// =====================================================================
extern "C" void kernel_launch(void* const* d_in, const int* in_sizes, int n_in,
                              void* d_out, int out_size, void* d_ws, size_t ws_size,
                              hipStream_t stream) {
    const float* x      = (const float*)d_in[0];
    const int*   ei     = (const int*)d_in[1];
    const int*   et     = (const int*)d_in[2];
    const float* basis[3] = {(const float*)d_in[3], (const float*)d_in[7],  (const float*)d_in[11]};
    const float* comp [3] = {(const float*)d_in[4], (const float*)d_in[8],  (const float*)d_in[12]};
    const float* root [3] = {(const float*)d_in[5], (const float*)d_in[9],  (const float*)d_in[13]};
    const float* bias [3] = {(const float*)d_in[6], (const float*)d_in[10], (const float*)d_in[14]};

    // workspace layout (floats); offsets all multiples of 4 floats -> 16B aligned
    float* w = (float*)d_ws;
    size_t off = 0;
    float* h      = w + off; off += (size_t)NREL * NN * FDIM;   // 5,120,000
    float* cnt    = w + off; off += (size_t)NREL * NN;          // 20,000
    float* inv    = w + off; off += (size_t)NREL * NN;          // 20,000
    float* Wcat   = w + off; off += (size_t)3 * FDIM * 256;     // 196,608 (max dpad)
    float* act1   = w + off; off += (size_t)NN * FDIM;          // 2,560,000
    float* act2   = w + off; off += (size_t)NN * FDIM;          // 2,560,000
    float* logits = w + off; off += (size_t)NN * 40;            // 400,000

    const long hElems = (long)NREL * NN * FDIM;

    // ---- graph-structure counts (same every layer) ----
    rgcn_zero_f<<<(NREL * NN + 255) / 256, 256, 0, stream>>>(cnt, NREL * NN);
    rgcn_count<<<(EE + 255) / 256, 256, 0, stream>>>(ei, et, cnt);
    rgcn_inv<<<(NREL * NN + 255) / 256, 256, 0, stream>>>(cnt, inv);

    const float* xin[3]  = {x, act1, act2};
    float*       xout[3] = {act1, act2, logits};
    const int    dout[3] = {256, 256, 40};
    const int    dpad[3] = {256, 256, 64};

    for (int l = 0; l < 3; ++l) {
        rgcn_zero_f<<<(int)((hElems + 255) / 256), 256, 0, stream>>>(h, hElems);
        rgcn_aggregate<<<(EE + 7) / 8, 256, 0, stream>>>(xin[l], ei, et, h);
        long wElems = (long)3 * FDIM * dpad[l];
        rgcn_build_w<<<(int)((wElems + 255) / 256), 256, 0, stream>>>(
            basis[l], comp[l], root[l], Wcat, dout[l], dpad[l]);
        dim3 grid((NN + TM - 1) / TM, dpad[l] / TN);
        rgcn_gemm_wmma<<<grid, 256, 0, stream>>>(
            xin[l], h, inv, Wcat, bias[l], xout[l], NN, dout[l], dpad[l], (l < 2) ? 1 : 0);
    }

    rgcn_logsoftmax<<<(NN + 7) / 8, 256, 0, stream>>>(logits, (float*)d_out);
}